// LogicLayer_79259326480585
// MI455X (gfx1250) — compile-verified
//
#include <hip/hip_runtime.h>
#include <hip/hip_bf16.h>
#include <stdint.h>

typedef float v2f __attribute__((ext_vector_type(2)));
typedef float v8f __attribute__((ext_vector_type(8)));

#define IN_DIM    8192
#define BATCH     512
#define OUT_DIM   65536
#define BJ        32      // output columns per block (main kernel)
#define COL_STRIDE 516    // floats per staged LDS column: 16B aligned, 2-way bank conflict only

// ------------------------------------------------------------------
// Kernel 1: x (512 x 8192) -> xT (8192 x 512), LDS-tiled transpose
// ------------------------------------------------------------------
__global__ void transpose_kernel(const float* __restrict__ x, float* __restrict__ xT) {
    __shared__ float tile[32][33];
    const int tx = threadIdx.x, ty = threadIdx.y;      // 32 x 8
    const int col0 = blockIdx.x * 32;                  // along in_dim
    const int row0 = blockIdx.y * 32;                  // along batch
#pragma unroll
    for (int i = 0; i < 32; i += 8)
        tile[ty + i][tx] = x[(size_t)(row0 + ty + i) * IN_DIM + (col0 + tx)];
    __syncthreads();
#pragma unroll
    for (int i = 0; i < 32; i += 8)
        xT[(size_t)(col0 + ty + i) * BATCH + (row0 + tx)] = tile[tx][ty + i];
}

// ------------------------------------------------------------------
// Kernel 2: k[j,0:4] = softmax(weights[j,:]) @ OP_COEFFS via
// v_wmma_f32_16x16x4_f32. One wave handles 16 neurons: P(16x16) @ OPC(16x4)
// as 4 chained K=4 WMMAs.
//  * softmax max/sum: pure reductions over float4 loads (no register arrays)
//  * lane-dependent A slots: re-loaded from memory at `row + 2*half`
//    (runtime ADDRESS, not runtime register index -> no cndmask trees)
//  * OPC generated in closed form: OPC[r][0]=h, [1]=f-h, [2]=g-h,
//    [3]=e-f-g+h with e=r&1,f=(r>>1)&1,g=(r>>2)&1,h=(r>>3)&1; for row
//    r = 4t+2*half+s only f=half is runtime.
// ------------------------------------------------------------------
__global__ void kcoef_kernel(const float* __restrict__ w, float* __restrict__ kbuf) {
    const int lane = threadIdx.x & 31;
    const int wave = (blockIdx.x * blockDim.x + threadIdx.x) >> 5;
    const int nb   = wave * 16;            // first neuron of this wave
    const int m    = lane & 15;            // matrix row / B column of this lane
    const int half = lane >> 4;            // 0: K=0/1 slots, 1: K=2/3 slots

    const float* row = w + (size_t)(nb + m) * 16;
    const float4* wr = (const float4*)row;
    const float4 q0 = wr[0], q1 = wr[1], q2 = wr[2], q3 = wr[3];

    // softmax denominator: max + sum of exps (reductions only)
    float mx = fmaxf(fmaxf(fmaxf(q0.x, q0.y), fmaxf(q0.z, q0.w)),
                     fmaxf(fmaxf(q1.x, q1.y), fmaxf(q1.z, q1.w)));
    mx = fmaxf(mx, fmaxf(fmaxf(fmaxf(q2.x, q2.y), fmaxf(q2.z, q2.w)),
                         fmaxf(fmaxf(q3.x, q3.y), fmaxf(q3.z, q3.w))));
    float s =
        __expf(q0.x - mx) + __expf(q0.y - mx) + __expf(q0.z - mx) + __expf(q0.w - mx) +
        __expf(q1.x - mx) + __expf(q1.y - mx) + __expf(q1.z - mx) + __expf(q1.w - mx) +
        __expf(q2.x - mx) + __expf(q2.y - mx) + __expf(q2.z - mx) + __expf(q2.w - mx) +
        __expf(q3.x - mx) + __expf(q3.y - mx) + __expf(q3.z - mx) + __expf(q3.w - mx);
    const float inv = 1.f / s;

    // the 8 logits this lane feeds into A: rows 4t + 2*half + s, s=0,1
    const float* rsel = row + 2 * half;          // 8-byte aligned either way
    const float2 r0 = *(const float2*)(rsel + 0);
    const float2 r1 = *(const float2*)(rsel + 4);
    const float2 r2 = *(const float2*)(rsel + 8);
    const float2 r3 = *(const float2*)(rsel + 12);

    const bool  c0b = (m & 1) != 0;        // B column bit 0
    const bool  c1b = (m & 2) != 0;        // B column bit 1
    const float cm  = (m < 4) ? 1.f : 0.f; // B columns 4..15 zero-padded
    const float fh  = (float)half;

    v8f acc = {};
#pragma unroll
    for (int t = 0; t < 4; ++t) {
        // A chunk: lanes 0-15: VGPR0=K0, VGPR1=K1 ; lanes 16-31: VGPR0=K2, VGPR1=K3
        const float2 rt = (t == 0) ? r0 : (t == 1) ? r1 : (t == 2) ? r2 : r3;
        v2f a, b;
        a.x = __expf(rt.x - mx) * inv;
        a.y = __expf(rt.y - mx) * inv;

        // B chunk: rows r = 4t + 2*half + s (s=0 -> b.x, s=1 -> b.y), col c=m.
        const float g = (float)(t & 1);
        const float h = (float)((t >> 1) & 1);
        const float col0  = h;                   // compile-time per t
        const float col1  = fh - h;              // single op on half
        const float col2  = g - h;               // compile-time per t
        const float col3x = 0.f - fh - g + h;    // s = 0
        const float col3y = 1.f - fh - g + h;    // s = 1
        const float vx01 = c0b ? col1 : col0;
        const float vx23 = c0b ? col3x : col2;
        const float vy23 = c0b ? col3y : col2;
        b.x = cm * (c1b ? vx23 : vx01);
        b.y = cm * (c1b ? vy23 : vx01);

        acc = __builtin_amdgcn_wmma_f32_16x16x4_f32(
            /*neg_a=*/false, a, /*neg_b=*/false, b,
            /*c_mod=*/(short)0, acc, /*reuse_a=*/false, /*reuse_b=*/false);
    }

    // D layout: VGPR v -> M=v (lanes 0-15) / M=v+8 (lanes 16-31), N=lane%16.
    if (m < 4) {
#pragma unroll
        for (int v = 0; v < 8; ++v)
            kbuf[(size_t)(nb + v + 8 * half) * 4 + (m & 3)] = acc[v];
    }
}

// ------------------------------------------------------------------
// Kernel 3: main gather + elementwise. Block = 32 output columns x 512 rows.
// Stage a/b source columns of xT into LDS with async global->LDS DMA,
// then emit coalesced nontemporal stores.
// ------------------------------------------------------------------
__global__ void logic_main_kernel(const float* __restrict__ xT,
                                  const float* __restrict__ kbuf,
                                  const long long* __restrict__ conn,
                                  float* __restrict__ out) {
    extern __shared__ float smem[];      // 64 columns * COL_STRIDE floats (dynamic)
    __shared__ int conn_s[64];           // block's gather indices
    const int tid = threadIdx.x;         // 256 threads = 8 waves
    const int j0  = blockIdx.x * BJ;

    // stage the 64 distinct connection indices once
    if (tid < 64)
        conn_s[tid] = (int)conn[(size_t)(j0 + (tid & 31)) * 2 + (tid >> 5)];
    __syncthreads();

    const uint32_t lds_base = (uint32_t)(uintptr_t)smem;  // low 32b of flat = LDS offset

    // ---- Phase A: async-stage 64 columns (32 a-srcs + 32 b-srcs), 2 KiB each.
    // 64 cols * 128 chunks(16B) = 8192 lane-transfers; 32 rounds over 256 lanes.
#pragma unroll 4
    for (int it = 0; it < 32; ++it) {
        const int t     = tid + it * 256;
        const int cidx  = t >> 7;           // 0..63  (staged column)
        const int chunk = t & 127;          // 16-byte chunk = 4 batch rows
        const int src   = conn_s[cidx];
        const float* g  = xT + (size_t)src * BATCH + chunk * 4;
        const uint32_t ldsaddr =
            lds_base + (uint32_t)(cidx * (COL_STRIDE * 4) + chunk * 16);
        asm volatile("global_load_async_to_lds_b128 %0, %1, off"
                     :: "v"(ldsaddr), "v"(g)
                     : "memory");
    }
    asm volatile("s_wait_asynccnt 0" ::: "memory");
    __syncthreads();

    // ---- Phase B: each wave covers one batch row x 32 consecutive columns
    // -> 128 B fully coalesced nontemporal stores.
    const int j  = tid & 31;
    const int rw = tid >> 5;                // 0..7
    const float4 kk = reinterpret_cast<const float4*>(kbuf)[j0 + j];
    const float* Acol = smem + (size_t)j * COL_STRIDE;
    const float* Bcol = smem + (size_t)(BJ + j) * COL_STRIDE;
    float* o = out + (size_t)j0 + j;
#pragma unroll 4
    for (int r = rw; r < BATCH; r += 8) {
        const float a = Acol[r];
        const float b = Bcol[r];
        const float v = kk.x + kk.y * a + kk.z * b + kk.w * (a * b);
        __builtin_nontemporal_store(v, o + (size_t)r * OUT_DIM);
    }
}

// ------------------------------------------------------------------
extern "C" void kernel_launch(void* const* d_in, const int* in_sizes, int n_in,
                              void* d_out, int out_size, void* d_ws, size_t ws_size,
                              hipStream_t stream) {
    const float* x        = (const float*)d_in[0];
    const float* weights  = (const float*)d_in[1];
    const long long* conn = (const long long*)d_in[2];
    float* out            = (float*)d_out;

    // workspace: xT (8192*512 f32 = 16 MiB) then kbuf (65536*4 f32 = 1 MiB)
    float* xT   = (float*)d_ws;
    float* kbuf = xT + (size_t)IN_DIM * BATCH;

    {   // transpose x -> xT
        dim3 tb(32, 8);
        dim3 tg(IN_DIM / 32, BATCH / 32);
        transpose_kernel<<<tg, tb, 0, stream>>>(x, xT);
    }
    {   // k coefficients via WMMA: 65536 neurons / 16 per wave / 8 waves per block
        kcoef_kernel<<<OUT_DIM / 16 / 8, 256, 0, stream>>>(weights, kbuf);
    }
    {   // main kernel: 2048 blocks x 256 threads, 129 KiB dynamic LDS
        size_t smem_bytes = (size_t)64 * COL_STRIDE * sizeof(float);
        logic_main_kernel<<<OUT_DIM / BJ, 256, smem_bytes, stream>>>(xT, kbuf, conn, out);
    }
}